// MVDR_oracle_using_steering_vector2_68135361183978
// MI455X (gfx1250) — compile-verified
//
#include <hip/hip_runtime.h>

typedef __attribute__((ext_vector_type(2))) float v2f;
typedef __attribute__((ext_vector_type(8))) float v8f;

#define NB   8     // batch
#define CCH  8     // channels
#define TT   600   // time frames
#define FF   513   // freq bins
#define WPB  8     // waves per block (8 * 32 = 256 threads)

// ---------------------------------------------------------------------------
// Kernel A: per-(n,f) spatial covariance via V_WMMA_F32_16X16X4_F32 + MVDR solve.
// One wave32 handles one (n,f). A/B operands are the SAME registers:
//   S[16][4] rows = [xr_0..xr_7, xi_0..xi_7], cols = 4 consecutive time steps.
//   D = S * S^T (16x16 fp32 accum) -> blocks [RR RI; IR II].
//   phi = (RR + II)/T + i*(IR - RI)/T  (+ LOAD*(1+i) on the diagonal)
// Writes conj(w)[n,f,c] (float2) into the workspace.
// ---------------------------------------------------------------------------
__global__ void mvdr_weights_kernel(const float* __restrict__ target,
                                    const float* __restrict__ sv,
                                    float* __restrict__ wgt) {
  __shared__ float lds[WPB][16][16];
  __shared__ float nume_s[WPB][CCH][2];

  const int tid  = threadIdx.x;
  const int wave = tid >> 5;
  const int lane = tid & 31;
  const int idx  = blockIdx.x * WPB + wave;   // flat n*F + f; N*F = 4104 = 513*8 exact
  const int n    = idx / FF;
  const int f    = idx - n * FF;

  // Per-lane source row: channel ch = lane%16 (0-7 real, 8-15 imag), K offset 0 or 2.
  const int ch   = lane & 15;
  const int ri   = ch >> 3;
  const int c0   = ch & 7;
  const int koff = (lane >> 4) << 1;

  const float* pt = target +
      (((size_t)((n * 2 + ri) * CCH + c0)) * TT + (size_t)koff) * FF + f;

  v8f acc = {};
  #pragma unroll 2
  for (int t0 = 0; t0 < TT; t0 += 4) {
    v2f a;
    a.x = pt[0];        // time t0 + koff
    a.y = pt[FF];       // time t0 + koff + 1
    pt += 4 * FF;
    // D += A * B with A == B (Gram matrix): v_wmma_f32_16x16x4_f32
    acc = __builtin_amdgcn_wmma_f32_16x16x4_f32(false, a, false, a,
                                                (short)0, acc, false, false);
  }

  // Spill 16x16 accumulator to LDS: VGPR v, lanes 0-15 -> row v, lanes 16-31 -> row v+8.
  #pragma unroll
  for (int v = 0; v < 8; ++v)
    lds[wave][v + ((lane >> 4) << 3)][lane & 15] = acc[v];

  __syncthreads();

  const float LOADF = 7.0710678118654755e-04f;  // 0.001 / sqrt(2)
  const float invT  = 1.0f / (float)TT;

  if (lane < CCH) {
    const int c = lane;
    float pr[8], pi[8], dr[8], di[8];
    #pragma unroll
    for (int e = 0; e < 8; ++e) {
      pr[e] = (lds[wave][c][e]     + lds[wave][c + 8][e + 8]) * invT;  // RR + II
      pi[e] = (lds[wave][c + 8][e] - lds[wave][c][e + 8])     * invT;  // IR - RI
      dr[e] = sv[(size_t)f * CCH + e];                                  // sv[0,f,e,0]
      di[e] = sv[(size_t)FF * CCH + (size_t)f * CCH + e];               // sv[1,f,e,0]
    }
    pr[c] += LOADF;   // complex diagonal loading LOAD*(1+i)
    pi[c] += LOADF;

    float nr = 0.f, ni = 0.f;                    // nume[c] = sum_e phi[c][e] * d[e]
    #pragma unroll
    for (int e = 0; e < 8; ++e) {
      nr += pr[e] * dr[e] - pi[e] * di[e];
      ni += pr[e] * di[e] + pi[e] * dr[e];
    }
    nume_s[wave][c][0] = nr;
    nume_s[wave][c][1] = ni;
  }

  __syncthreads();

  if (lane < CCH) {
    const int c = lane;
    float der = 0.f, dei = 0.f;                  // deno = sum_c conj(d[c]) * nume[c]
    #pragma unroll
    for (int cc = 0; cc < 8; ++cc) {
      float xr  = nume_s[wave][cc][0];
      float xi  = nume_s[wave][cc][1];
      float ddr = sv[(size_t)f * CCH + cc];
      float ddi = sv[(size_t)FF * CCH + (size_t)f * CCH + cc];
      der += ddr * xr + ddi * xi;
      dei += ddr * xi - ddi * xr;
    }
    const float nr  = nume_s[wave][c][0];
    const float ni  = nume_s[wave][c][1];
    const float inv = 1.0f / (der * der + dei * dei);
    const float wr  = (nr * der + ni * dei) * inv;   // w = nume * conj(deno)/|deno|^2
    const float wi  = (ni * der - nr * dei) * inv;
    float2 cw;
    cw.x = wr;
    cw.y = -wi;                                      // store conj(w)
    ((float2*)wgt)[(size_t)idx * CCH + c] = cw;
  }
}

// ---------------------------------------------------------------------------
// Kernel B: beamform  x_bf[n,f,t] = sum_c conj(w)[n,f,c] * y[n,c,t,f].
// Pure streaming, coalesced along f. Output layout [N,2,1,T,F].
// ---------------------------------------------------------------------------
__global__ void mvdr_beamform_kernel(const float* __restrict__ mixture,
                                     const float* __restrict__ wgt,
                                     float* __restrict__ out) {
  const int total = NB * TT * FF;
  const int idx = blockIdx.x * blockDim.x + threadIdx.x;
  if (idx >= total) return;
  const int f  = idx % FF;
  const int nt = idx / FF;
  const int t  = nt % TT;
  const int n  = nt / TT;

  const float2* cw = (const float2*)wgt + (size_t)(n * FF + f) * CCH;
  const size_t plane = (size_t)TT * FF;
  const float* yr = mixture + (size_t)(n * 2 + 0) * CCH * plane + (size_t)t * FF + f;
  const float* yi = mixture + (size_t)(n * 2 + 1) * CCH * plane + (size_t)t * FF + f;

  float ar = 0.f, ai = 0.f;
  #pragma unroll
  for (int c = 0; c < CCH; ++c) {
    const float2 w = cw[c];                // (Re conj(w), Im conj(w))
    const float r = yr[(size_t)c * plane];
    const float i = yi[(size_t)c * plane];
    ar += w.x * r - w.y * i;
    ai += w.x * i + w.y * r;
  }
  out[((size_t)(n * 2 + 0) * TT + t) * FF + f] = ar;
  out[((size_t)(n * 2 + 1) * TT + t) * FF + f] = ai;
}

extern "C" void kernel_launch(void* const* d_in, const int* in_sizes, int n_in,
                              void* d_out, int out_size, void* d_ws, size_t ws_size,
                              hipStream_t stream) {
  const float* mixture = (const float*)d_in[0];
  // d_in[1] (noise) is unused by the reference computation.
  const float* target  = (const float*)d_in[2];
  const float* sv      = (const float*)d_in[3];
  float* out = (float*)d_out;
  float* wgt = (float*)d_ws;   // needs N*F*C*2 floats = 262,656 bytes

  (void)in_sizes; (void)n_in; (void)out_size; (void)ws_size;

  const int nf = NB * FF;                    // 4104 (n,f) pairs, one wave each
  mvdr_weights_kernel<<<dim3(nf / WPB), dim3(WPB * 32), 0, stream>>>(target, sv, wgt);

  const int total = NB * TT * FF;
  mvdr_beamform_kernel<<<dim3((total + 255) / 256), dim3(256), 0, stream>>>(mixture, wgt, out);
}